// MultiValuesWeightedAttentionPooling_12386685682158
// MI455X (gfx1250) — compile-verified
//
#include <hip/hip_runtime.h>
#include <hip/hip_bf16.h>

// ---------------- problem constants ----------------
#define B 32
#define S 2048
#define DIN 1024
#define DOUT 768
#define NV 10
#define NEGV (-1.0e9f)
#define SC 32               // S chunks for pass-1 two-stage reduction
#define SROWS (S / SC)      // 64 rows per chunk
#define CCH 64              // chunks in fused pass 2 (32 rows each)

typedef __attribute__((ext_vector_type(2))) float v2f;
typedef __attribute__((ext_vector_type(8))) float v8f;

// ---------------- WMMA f32 GEMM: D[M,N] = A[M,K] @ B(^T) + rowScale*bias ----------------
// one wave (32 threads) per 16x16 output tile; K multiple of 4
template <int TRANSB>
__global__ __launch_bounds__(32) void wmma_gemm_f32(
    const float* __restrict__ A, int lda,
    const float* __restrict__ Bm, int ldb,
    float* __restrict__ D, int ldd, int K,
    const float* __restrict__ bias, const float* __restrict__ rowScale)
{
    int lane = threadIdx.x;
    int half = lane >> 4;          // 0: lanes 0-15, 1: lanes 16-31
    int lm   = lane & 15;
    int m0 = blockIdx.y * 16;
    int n0 = blockIdx.x * 16;

    // A row for this lane: element K index = k0 + 2*half + {0,1} -> contiguous float2
    const float2* A2 = (const float2*)(A + (size_t)(m0 + lm) * lda);
    const float2* Bc2 = TRANSB ? (const float2*)(Bm + (size_t)(n0 + lm) * ldb) : nullptr;
    const float*  Br  = TRANSB ? nullptr : (Bm + n0 + lm);

    v8f acc = {0.f, 0.f, 0.f, 0.f, 0.f, 0.f, 0.f, 0.f};

#pragma unroll 8
    for (int k0 = 0; k0 < K; k0 += 4) {
        int ka = k0 + 2 * half;
        float2 at = A2[(k0 >> 1) + half];
        v2f a; a.x = at.x; a.y = at.y;
        v2f b;
        if (TRANSB) {
            float2 bt = Bc2[(k0 >> 1) + half];
            b.x = bt.x; b.y = bt.y;
        } else {
            b.x = Br[(size_t)ka * ldb];
            b.y = Br[(size_t)(ka + 1) * ldb];
        }
        acc = __builtin_amdgcn_wmma_f32_16x16x4_f32(
            false, a, false, b, (short)0, acc, false, false);
    }

    int n = n0 + lm;
#pragma unroll
    for (int r = 0; r < 8; r++) {
        int m = m0 + r + 8 * half;
        float v = acc[r];
        if (bias) v += (rowScale ? rowScale[m] : 1.0f) * bias[n];
        D[(size_t)m * ldd + n] = v;
    }
}

// ---------------- build float masks: wf = em*frame, wp = em*post ----------------
__global__ void make_weights(const int* __restrict__ em, const int* __restrict__ pmask,
                             const int* __restrict__ fmask,
                             float* __restrict__ wp, float* __restrict__ wf)
{
    int i = blockIdx.x * blockDim.x + threadIdx.x;
    if (i < B * S) {
        float e = (float)em[i];
        wp[i] = e * (float)pmask[i];
        wf[i] = e * (float)fmask[i];
    }
}

// ---------------- pass 1: masked column-sum over S (float4, per-chunk partials) -------
// grid (B, SC), block 256; p1[(chunk*B + b)*256 + t] (float4)
__global__ __launch_bounds__(256) void wcolsum4(
    const float4* __restrict__ emb4, const float* __restrict__ w1,
    float4* __restrict__ p1)
{
    int b = blockIdx.x, ch = blockIdx.y, t = threadIdx.x;
    const float4* e  = emb4 + ((size_t)b * S + (size_t)ch * SROWS) * (DIN / 4) + t;
    const float*  a1 = w1 + (size_t)b * S + (size_t)ch * SROWS;
    float4 acc; acc.x = acc.y = acc.z = acc.w = 0.f;
    for (int s = 0; s < SROWS; s++) {
        float w = a1[s];
        float4 x = e[(size_t)s * (DIN / 4)];
        acc.x += w * x.x; acc.y += w * x.y; acc.z += w * x.z; acc.w += w * x.w;
    }
    p1[((size_t)ch * B + b) * (DIN / 4) + t] = acc;
}

__global__ __launch_bounds__(256) void combine4(const float4* __restrict__ part,
                                                float4* __restrict__ outv)
{
    int b = blockIdx.x, t = threadIdx.x;
    float4 acc; acc.x = acc.y = acc.z = acc.w = 0.f;
    for (int c = 0; c < SC; c++) {
        float4 x = part[((size_t)c * B + b) * (DIN / 4) + t];
        acc.x += x.x; acc.y += x.y; acc.z += x.z; acc.w += x.w;
    }
    outv[(size_t)b * (DIN / 4) + t] = acc;
}

// count_f[b] = sum_s wf[b,s]
__global__ void rowsum_s(const float* __restrict__ w, float* __restrict__ outv)
{
    __shared__ float red[256];
    int b = blockIdx.x;
    float a = 0.f;
    for (int i = threadIdx.x; i < S; i += 256) a += w[(size_t)b * S + i];
    red[threadIdx.x] = a; __syncthreads();
    for (int off = 128; off > 0; off >>= 1) {
        if (threadIdx.x < off) red[threadIdx.x] += red[threadIdx.x + off];
        __syncthreads();
    }
    if (threadIdx.x == 0) outv[b] = red[0];
}

// ---------------- tiny value attention over Nv=10 ----------------
__global__ void value_attn_kernel(const float* __restrict__ ef, const float* __restrict__ V,
                                  const int* __restrict__ vmask, const float* __restrict__ bias,
                                  float* __restrict__ vemb_out, float* __restrict__ vprobs_out,
                                  float* __restrict__ c_out)
{
    __shared__ float red[256];
    __shared__ float scv[NV];
    __shared__ float pr[NV];
    int b = blockIdx.x;
    const float* e = ef + (size_t)b * DOUT;

    for (int v = 0; v < NV; v++) {
        float p = 0.f;
        for (int d = threadIdx.x; d < DOUT; d += 256) p += e[d] * V[(size_t)v * DOUT + d];
        red[threadIdx.x] = p; __syncthreads();
        for (int off = 128; off > 0; off >>= 1) {
            if (threadIdx.x < off) red[threadIdx.x] += red[threadIdx.x + off];
            __syncthreads();
        }
        if (threadIdx.x == 0)
            scv[v] = red[0] + (1.0f - (float)vmask[b * NV + v]) * NEGV;
        __syncthreads();
    }
    if (threadIdx.x == 0) {
        float m = -3.0e38f;
        for (int v = 0; v < NV; v++) m = fmaxf(m, scv[v]);
        float s = 0.f;
        for (int v = 0; v < NV; v++) { pr[v] = expf(scv[v] - m); s += pr[v]; }
        float inv = 1.0f / s;
        for (int v = 0; v < NV; v++) pr[v] *= inv;
    }
    __syncthreads();

    float cp = 0.f;
    for (int d = threadIdx.x; d < DOUT; d += 256) {
        float acc = 0.f;
        for (int v = 0; v < NV; v++) acc += pr[v] * V[(size_t)v * DOUT + d];
        vemb_out[(size_t)b * DOUT + d] = acc;
        cp += bias[d] * acc;
    }
    red[threadIdx.x] = cp; __syncthreads();
    for (int off = 128; off > 0; off >>= 1) {
        if (threadIdx.x < off) red[threadIdx.x] += red[threadIdx.x + off];
        __syncthreads();
    }
    if (threadIdx.x == 0) c_out[b] = red[0];
    if (threadIdx.x < NV) vprobs_out[b * NV + threadIdx.x] = pr[threadIdx.x];
}

// ---------------- fused pass 2: scores + online-softmax weighted colsum ----------------
// grid (B, CCH=64 chunks of 32 rows), block 256 (8 waves x 4 rows).
// Outputs: raw scores sc[2*B*S], per-chunk (M,L) ml[2*B*CCH*2], qpart[2*B*CCH*DIN]
__global__ __launch_bounds__(256) void fused_pass2(
    const float4* __restrict__ emb4,
    const float4* __restrict__ u_c4, const float4* __restrict__ u_m4,
    const float* __restrict__ c_c, const float* __restrict__ c_m,
    const float* __restrict__ wp,
    float* __restrict__ sc, float* __restrict__ ml, float* __restrict__ qpart)
{
    __shared__ float4 suc[DIN / 4];
    __shared__ float4 sum_[DIN / 4];
    __shared__ float4 qbc[DIN / 4];
    __shared__ float4 qbm[DIN / 4];
    __shared__ float wmax[2][8], wsum[2][8];

    int b = blockIdx.x, chunk = blockIdx.y;
    int tid = threadIdx.x, wave = tid >> 5, lane = tid & 31;

    suc[tid]  = u_c4[(size_t)b * (DIN / 4) + tid];
    sum_[tid] = u_m4[(size_t)b * (DIN / 4) + tid];
    float4 z; z.x = z.y = z.z = z.w = 0.f;
    qbc[tid] = z; qbm[tid] = z;
    __syncthreads();

    float cc = c_c[b], cm = c_m[b];
    float m_c = -3.0e38f, l_c = 0.f, m_m = -3.0e38f, l_m = 0.f;
    float4 qa_c[8], qa_m[8];
#pragma unroll
    for (int j = 0; j < 8; j++) { qa_c[j] = z; qa_m[j] = z; }

    for (int r = 0; r < 4; r++) {
        int s = chunk * 32 + wave * 4 + r;
        const float4* row = emb4 + ((size_t)b * S + s) * (DIN / 4);
        float4 x[8];
#pragma unroll
        for (int j = 0; j < 8; j++) x[j] = row[lane + j * 32];

        float dc = 0.f, dm = 0.f;
#pragma unroll
        for (int j = 0; j < 8; j++) {
            float4 u1 = suc[lane + j * 32];
            float4 u2 = sum_[lane + j * 32];
            dc += x[j].x * u1.x + x[j].y * u1.y + x[j].z * u1.z + x[j].w * u1.w;
            dm += x[j].x * u2.x + x[j].y * u2.y + x[j].z * u2.z + x[j].w * u2.w;
        }
#pragma unroll
        for (int off = 16; off > 0; off >>= 1) {
            dc += __shfl_xor(dc, off, 32);
            dm += __shfl_xor(dm, off, 32);
        }
        float pen = (1.0f - wp[(size_t)b * S + s]) * NEGV;
        float svc = dc + cc + pen;
        float svm = dm + cm + pen;
        if (lane == 0) {
            sc[(size_t)(0 * B + b) * S + s] = svc;
            sc[(size_t)(1 * B + b) * S + s] = svm;
        }
        // online softmax update (cultural)
        {
            float nm = fmaxf(m_c, svc);
            float scl = expf(m_c - nm);
            float w = expf(svc - nm);
            l_c = l_c * scl + w;
#pragma unroll
            for (int j = 0; j < 8; j++) {
                qa_c[j].x = qa_c[j].x * scl + w * x[j].x;
                qa_c[j].y = qa_c[j].y * scl + w * x[j].y;
                qa_c[j].z = qa_c[j].z * scl + w * x[j].z;
                qa_c[j].w = qa_c[j].w * scl + w * x[j].w;
            }
            m_c = nm;
        }
        // online softmax update (moral)
        {
            float nm = fmaxf(m_m, svm);
            float scl = expf(m_m - nm);
            float w = expf(svm - nm);
            l_m = l_m * scl + w;
#pragma unroll
            for (int j = 0; j < 8; j++) {
                qa_m[j].x = qa_m[j].x * scl + w * x[j].x;
                qa_m[j].y = qa_m[j].y * scl + w * x[j].y;
                qa_m[j].z = qa_m[j].z * scl + w * x[j].z;
                qa_m[j].w = qa_m[j].w * scl + w * x[j].w;
            }
            m_m = nm;
        }
    }

    if (lane == 0) {
        wmax[0][wave] = m_c; wsum[0][wave] = l_c;
        wmax[1][wave] = m_m; wsum[1][wave] = l_m;
    }
    __syncthreads();

    float Mc = -3.0e38f, Mm = -3.0e38f;
#pragma unroll
    for (int i = 0; i < 8; i++) {
        Mc = fmaxf(Mc, wmax[0][i]);
        Mm = fmaxf(Mm, wmax[1][i]);
    }
    float ac = expf(m_c - Mc);  // uniform within wave
    float am = expf(m_m - Mm);

    // deterministic serialized accumulation of wave partials into LDS
    for (int w2 = 0; w2 < 8; w2++) {
        if (wave == w2) {
#pragma unroll
            for (int j = 0; j < 8; j++) {
                int idx = lane + j * 32;
                float4 t = qbc[idx];
                t.x += ac * qa_c[j].x; t.y += ac * qa_c[j].y;
                t.z += ac * qa_c[j].z; t.w += ac * qa_c[j].w;
                qbc[idx] = t;
                float4 u = qbm[idx];
                u.x += am * qa_m[j].x; u.y += am * qa_m[j].y;
                u.z += am * qa_m[j].z; u.w += am * qa_m[j].w;
                qbm[idx] = u;
            }
        }
        __syncthreads();
    }

    if (tid == 0) {
        float Lc = 0.f, Lm = 0.f;
        for (int i = 0; i < 8; i++) {
            Lc += expf(wmax[0][i] - Mc) * wsum[0][i];
            Lm += expf(wmax[1][i] - Mm) * wsum[1][i];
        }
        size_t mc0 = ((size_t)(0 * B + b) * CCH + chunk) * 2;
        size_t mm0 = ((size_t)(1 * B + b) * CCH + chunk) * 2;
        ml[mc0] = Mc; ml[mc0 + 1] = Lc;
        ml[mm0] = Mm; ml[mm0 + 1] = Lm;
    }
    float4* qp_c = (float4*)qpart + ((size_t)(0 * B + b) * CCH + chunk) * (DIN / 4);
    float4* qp_m = (float4*)qpart + ((size_t)(1 * B + b) * CCH + chunk) * (DIN / 4);
    qp_c[tid] = qbc[tid];
    qp_m[tid] = qbm[tid];
}

// ---------------- global softmax stats from chunk (M,L) ----------------
// grid 2*B, block 64 (one thread per chunk)
__global__ void softmax_stats(const float* __restrict__ ml, float* __restrict__ MgLg)
{
    __shared__ float sm[64], sl[64];
    int cb = blockIdx.x, t = threadIdx.x;
    float m = ml[((size_t)cb * CCH + t) * 2];
    float l = ml[((size_t)cb * CCH + t) * 2 + 1];
    sm[t] = m; __syncthreads();
    for (int off = 32; off > 0; off >>= 1) {
        if (t < off) sm[t] = fmaxf(sm[t], sm[t + off]);
        __syncthreads();
    }
    float Mg = sm[0]; __syncthreads();
    sl[t] = expf(m - Mg) * l; __syncthreads();
    for (int off = 32; off > 0; off >>= 1) {
        if (t < off) sl[t] += sl[t + off];
        __syncthreads();
    }
    if (t == 0) { MgLg[cb * 2] = Mg; MgLg[cb * 2 + 1] = sl[0]; }
}

// probs[cb, s] = exp(sc - Mg)/Lg ; grid 2*B
__global__ void probs_kernel(const float* __restrict__ sc, const float* __restrict__ MgLg,
                             float* __restrict__ probs)
{
    int cb = blockIdx.x;
    float Mg = MgLg[cb * 2];
    float inv = 1.0f / MgLg[cb * 2 + 1];
    for (int i = threadIdx.x; i < S; i += 256)
        probs[(size_t)cb * S + i] = expf(sc[(size_t)cb * S + i] - Mg) * inv;
}

// q[cb, d] = sum_i exp(m_i - Mg)*qpart_i[d] / Lg ; grid 2*B, block 256 (float4)
__global__ __launch_bounds__(256) void q_combine(
    const float4* __restrict__ qpart4, const float* __restrict__ ml,
    const float* __restrict__ MgLg, float4* __restrict__ q4)
{
    __shared__ float wgt[CCH];
    int cb = blockIdx.x, tid = threadIdx.x;
    float Mg = MgLg[cb * 2];
    float inv = 1.0f / MgLg[cb * 2 + 1];
    if (tid < CCH) wgt[tid] = expf(ml[((size_t)cb * CCH + tid) * 2] - Mg);
    __syncthreads();
    float4 acc; acc.x = acc.y = acc.z = acc.w = 0.f;
    const float4* base = qpart4 + (size_t)cb * CCH * (DIN / 4) + tid;
    for (int i = 0; i < CCH; i++) {
        float w = wgt[i];
        float4 t = base[(size_t)i * (DIN / 4)];
        acc.x += w * t.x; acc.y += w * t.y; acc.z += w * t.z; acc.w += w * t.w;
    }
    acc.x *= inv; acc.y *= inv; acc.z *= inv; acc.w *= inv;
    q4[(size_t)cb * (DIN / 4) + tid] = acc;
}

__global__ void vec_add(const float* __restrict__ a, const float* __restrict__ b2,
                        float* __restrict__ o, int n)
{
    int i = blockIdx.x * blockDim.x + threadIdx.x;
    if (i < n) o[i] = a[i] + b2[i];
}

// ---------------- host launch ----------------
extern "C" void kernel_launch(void* const* d_in, const int* in_sizes, int n_in,
                              void* d_out, int out_size, void* d_ws, size_t ws_size,
                              hipStream_t stream)
{
    const float* emb   = (const float*)d_in[0];
    const int*   em    = (const int*)d_in[1];
    const int*   cmask = (const int*)d_in[2];
    const int*   mmask = (const int*)d_in[3];
    const int*   pmask = (const int*)d_in[4];
    const int*   fmask = (const int*)d_in[5];
    const float* W     = (const float*)d_in[6];
    const float* bias  = (const float*)d_in[7];
    const float* Vc    = (const float*)d_in[8];
    const float* Vm    = (const float*)d_in[9];
    float* out = (float*)d_out;
    float* ws  = (float*)d_ws;

    // output offsets (flat, in return order)
    const size_t o_fc  = 0;
    const size_t o_fm  = o_fc  + (size_t)B * DOUT;
    const size_t o_fcp = o_fm  + (size_t)B * DOUT;
    const size_t o_fmp = o_fcp + (size_t)B * NV;
    const size_t o_pc  = o_fmp + (size_t)B * NV;
    const size_t o_pm  = o_pc  + (size_t)B * DOUT;   // contiguous with o_pc
    const size_t o_pcp = o_pm  + (size_t)B * DOUT;
    const size_t o_pmp = o_pcp + (size_t)B * S;      // contiguous with o_pcp
    const size_t o_pool= o_pmp + (size_t)B * S;

    // workspace layout (floats)
    float* wf      = ws;                                   // B*S
    float* wp      = wf + (size_t)B * S;                   // B*S
    float* qpart   = wp + (size_t)B * S;                   // 2*B*CCH*DIN (pass-2)
    float* part1   = qpart;                                // alias: pass-1 partials (SC*B*DIN fits)
    float* ebar    = qpart + (size_t)2 * B * CCH * DIN;    // B*DIN
    float* count_f = ebar + (size_t)B * DIN;               // B
    float* ef      = count_f + B;                          // B*DOUT
    float* c_c     = ef + (size_t)B * DOUT;                // B
    float* c_m     = c_c + B;                              // B
    float* u_c     = c_m + B;                              // B*DIN
    float* u_m     = u_c + (size_t)B * DIN;                // B*DIN
    float* sc      = u_m + (size_t)B * DIN;                // 2*B*S
    float* ml      = sc + (size_t)2 * B * S;               // 2*B*CCH*2
    float* MgLg    = ml + (size_t)2 * B * CCH * 2;         // 2*B*2
    float* q       = MgLg + (size_t)2 * B * 2;             // 2*B*DIN

    // 1) masks -> float weights
    make_weights<<<(B * S + 255) / 256, 256, 0, stream>>>(em, pmask, fmask, wp, wf);

    // 2) pass 1: frame-masked colsum of embeddings (two-stage)
    wcolsum4<<<dim3(B, SC), 256, 0, stream>>>((const float4*)emb, wf, (float4*)part1);
    combine4<<<B, 256, 0, stream>>>((const float4*)part1, (float4*)ebar);
    rowsum_s<<<B, 256, 0, stream>>>(wf, count_f);

    // 3) ef = ebar @ W + count_f * bias      [32,768]
    wmma_gemm_f32<0><<<dim3(DOUT / 16, B / 16), 32, 0, stream>>>(
        ebar, DIN, W, DOUT, ef, DOUT, DIN, bias, count_f);

    // 4) value attention -> frame outputs, vemb, c = bias.vemb
    value_attn_kernel<<<B, 256, 0, stream>>>(ef, Vc, cmask, bias, out + o_fc, out + o_fcp, c_c);
    value_attn_kernel<<<B, 256, 0, stream>>>(ef, Vm, mmask, bias, out + o_fm, out + o_fmp, c_m);

    // 5) u = vemb @ W^T                      [32,1024]
    wmma_gemm_f32<1><<<dim3(DIN / 16, B / 16), 32, 0, stream>>>(
        out + o_fc, DOUT, W, DOUT, u_c, DIN, DOUT, nullptr, nullptr);
    wmma_gemm_f32<1><<<dim3(DIN / 16, B / 16), 32, 0, stream>>>(
        out + o_fm, DOUT, W, DOUT, u_m, DIN, DOUT, nullptr, nullptr);

    // 6) fused pass 2: scores + online-softmax weighted colsum (single read of emb)
    fused_pass2<<<dim3(B, CCH), 256, 0, stream>>>(
        (const float4*)emb, (const float4*)u_c, (const float4*)u_m,
        c_c, c_m, wp, sc, ml, qpart);

    // 7) global softmax stats, prob outputs, q combine
    softmax_stats<<<2 * B, 64, 0, stream>>>(ml, MgLg);
    probs_kernel<<<2 * B, 256, 0, stream>>>(sc, MgLg, out + o_pcp);
    q_combine<<<2 * B, 256, 0, stream>>>((const float4*)qpart, ml, MgLg, (float4*)q);

    // 8) [q_c; q_m] @ W + bias -> post_cultural | post_moral (contiguous in d_out)
    wmma_gemm_f32<0><<<dim3(DOUT / 16, (2 * B) / 16), 32, 0, stream>>>(
        q, DIN, W, DOUT, out + o_pc, DOUT, DIN, bias, nullptr);

    // 9) pooled = post_cultural + post_moral
    vec_add<<<(B * DOUT + 255) / 256, 256, 0, stream>>>(
        out + o_pc, out + o_pm, out + o_pool, B * DOUT);
}